// dynamic_gcn_77876347011098
// MI455X (gfx1250) — compile-verified
//
#include <hip/hip_runtime.h>
#include <hip/hip_bf16.h>

// ---- problem constants (match reference) ----
#define B_DIM 16
#define C_DIM 32      // input channels
#define N_DIM 128     // nodes
#define T_DIM 12      // timesteps
#define CO_DIM 64     // output channels
#define ROW (N_DIM * T_DIM)   // 1536 floats = one (k,l) slab / one A j-row

typedef __attribute__((ext_vector_type(2))) float v2f;
typedef __attribute__((ext_vector_type(4))) float v4f;
typedef __attribute__((ext_vector_type(8))) float v8f;

// =====================================================================
// Kernel 1: dynamic graph propagation (memory-bound: streams 402 MB of A)
//   x1[k,l] = sum_j x[j,l]  * A[j,k,l]      (per b,c)
//   x2[k,l] = sum_j x1[j,l] * A[j,k,l]
// One block per (b,c). A block = 128 rows x 1536 contiguous floats.
// 384 threads x 4 floats = one row exactly: per thread per j-row this is
// ONE global_load_b128 + ONE ds_load_b128 + 4 FMAs (4t%12 in {0,4,8}, so
// the 4 elements share k, span l..l+3 with no wrap, and the xs read is
// 16B-aligned -> merges to b128).
// =====================================================================
__global__ __launch_bounds__(384)
void prop_kernel(const float* __restrict__ x,
                 const float* __restrict__ A,
                 float* __restrict__ h)  // h: [B, 2C, N, T]
{
    __shared__ float xs[ROW];    // x slab for this (b,c)
    __shared__ float x1s[ROW];   // order-1 result, reused as pass-2 vector

    const int t  = threadIdx.x;          // [0,384)
    const int m0 = 4 * t;                // this thread's 4 elements: m0..m0+3
    const int l0 = m0 % T_DIM;           // in {0,4,8}; l0+3 <= 11 (no wrap)
    const int bc = blockIdx.x;           // b*C + c
    const int b  = bc >> 5;
    const int c  = bc & 31;

    const float* xg = x + (size_t)bc * ROW;
    const float* Ab = A + (size_t)bc * N_DIM * ROW;

    // stage x slab (one b128 per thread)
    *(v4f*)(xs + m0) = *(const v4f*)(xg + m0);
    __syncthreads();

    // ---- pass 1: x1 = x^T A (A streamed from HBM once, b128 per row) ----
    v4f acc = {};
#pragma unroll 4
    for (int j = 0; j < N_DIM; ++j) {
        const v4f av = *(const v4f*)(Ab + (size_t)j * ROW + m0);
        const v4f xv = *(const v4f*)(xs + j * T_DIM + l0);   // 16B aligned
#pragma unroll
        for (int i = 0; i < 4; ++i)
            acc[i] += xv[i] * av[i];
    }
    float* h1 = h + ((size_t)b * (2 * C_DIM) + c) * ROW;
    *(v4f*)(h1 + m0)  = acc;
    *(v4f*)(x1s + m0) = acc;
    __syncthreads();

    // ---- pass 2: x2 = x1^T A. Same 786 KB re-read (ideally L2 hits from
    // pass 1); NT hint = last pass over these lines, don't re-prioritize. ----
    v4f acc2 = {};
#pragma unroll 4
    for (int j = 0; j < N_DIM; ++j) {
        const v4f av = __builtin_nontemporal_load(
            (const v4f*)(Ab + (size_t)j * ROW + m0));
        const v4f xv = *(const v4f*)(x1s + j * T_DIM + l0);
#pragma unroll
        for (int i = 0; i < 4; ++i)
            acc2[i] += xv[i] * av[i];
    }
    float* h2 = h + ((size_t)b * (2 * C_DIM) + C_DIM + c) * ROW;
    *(v4f*)(h2 + m0) = acc2;
}

// =====================================================================
// Kernel 2: 1x1 conv == GEMM  out[b] = W[64x64] * h[b][64x1536] + bias
// fp32 WMMA: V_WMMA_F32_16X16X4_F32, one 16x16 tile per wave.
//   M = output channel (4 tiles), Ncol = n*T flat (96 tiles), K = 64 (16 steps)
// 6144 tiles total = 768 blocks * 8 waves, exactly one tile per wave
// (no divergence -> EXEC all ones around WMMA, as the ISA requires).
// =====================================================================
__global__ __launch_bounds__(256)
void mix_kernel(const float* __restrict__ h,
                const float* __restrict__ W,     // [CO_DIM, 2C] row-major
                const float* __restrict__ bias,  // [CO_DIM]
                float* __restrict__ out)         // [B, CO_DIM, N, T]
{
    const int lane = threadIdx.x & 31;
    const int wave = threadIdx.x >> 5;
    const int tile = blockIdx.x * 8 + wave;      // [0, 6144)

    const int b   = tile / 384;
    const int rem = tile - b * 384;
    const int ot  = rem / 96;                    // output-channel tile
    const int mt  = rem - ot * 96;               // (n,l) tile
    const int o0  = ot * 16;
    const int m0  = mt * 16;

    const int lm = lane & 15;                    // M (for A) / Ncol (for B,C,D)
    const int lg = lane >> 4;                    // lane group

    const float* hb = h + (size_t)b * (2 * C_DIM) * ROW;

    v8f acc = {};
#pragma unroll 4
    for (int k0 = 0; k0 < 2 * C_DIM; k0 += 4) {
        // A fragment: W tile rows o0..o0+15, cols k0..k0+3
        //   lane m=lm, VGPR0/1 hold K = 2*lg, 2*lg+1  (ISA 32-bit A 16x4 layout)
        const float* wrow = W + (size_t)(o0 + lm) * (2 * C_DIM) + k0 + 2 * lg;
        v2f a;
        a.x = wrow[0];
        a.y = wrow[1];
        // B fragment: h tile rows (channels) k0..k0+3, cols m0..m0+15
        //   VGPR r, lane-group g -> K = r + 2*g  (row-striped-across-lanes layout)
        const float* hcol = hb + (size_t)(k0 + 2 * lg) * ROW + m0 + lm;
        v2f bb;
        bb.x = hcol[0];
        bb.y = hcol[ROW];
        // 8 args: (neg_a, A, neg_b, B, c_mod, C, reuse_a, reuse_b)
        acc = __builtin_amdgcn_wmma_f32_16x16x4_f32(
            false, a, false, bb, (short)0, acc, false, false);
    }

    // C/D layout: VGPR r -> M = r + 8*lg, lane lm -> Ncol
    float* ob = out + (size_t)b * CO_DIM * ROW + m0 + lm;
#pragma unroll
    for (int r = 0; r < 8; ++r) {
        const int o = o0 + r + 8 * lg;
        ob[(size_t)o * ROW] = acc[r] + bias[o];
    }
}

extern "C" void kernel_launch(void* const* d_in, const int* in_sizes, int n_in,
                              void* d_out, int out_size, void* d_ws, size_t ws_size,
                              hipStream_t stream)
{
    const float* x    = (const float*)d_in[0];  // [B, C, N, T]
    const float* A    = (const float*)d_in[1];  // [B, C, N, N, T]
    const float* W    = (const float*)d_in[2];  // [CO, 2C]
    const float* bias = (const float*)d_in[3];  // [CO]
    float* out = (float*)d_out;                 // [B, CO, N, T]
    float* h   = (float*)d_ws;                  // [B, 2C, N, T] = 6 MB scratch

    // Pass A: propagation (512 blocks = one per (b,c), 384 threads = 12 waves)
    prop_kernel<<<B_DIM * C_DIM, 384, 0, stream>>>(x, A, h);
    // Pass B: channel-mix GEMM via fp32 WMMA (768 blocks * 8 waves = 6144 tiles)
    mix_kernel<<<768, 256, 0, stream>>>(h, W, bias, out);
}